// KalmanFormer_70884140253538
// MI455X (gfx1250) — compile-verified
//
#include <hip/hip_runtime.h>
#include <math.h>

// ---------------------------------------------------------------------------
// KalmanFormer for MI455X (gfx1250, wave32).
//   grid  = B (128) blocks, one batch element per workgroup (scan is batch-
//           independent, sequential only in T)
//   block = 256 threads = 8 waves; each wave owns one 16x16 WMMA output tile
//   All GEMMs use V_WMMA_F32_16X16X4_F32 (native fp32 matrix op) to match the
//   fp32 reference. Activations live in LDS (~50KB); weights (~0.5MB total)
//   stay L2-resident across the 128x128 step/batch reuse.
// ---------------------------------------------------------------------------

typedef float v2f __attribute__((ext_vector_type(2)));
typedef float v8f __attribute__((ext_vector_type(8)));

#define WIN  20
#define SEQP 32
#define Dm   64
#define Bsz  128
#define Tsz  128
#define KF_EPS 1e-5f

struct AttnP { const float *wq,*wk,*wv,*wo,*bq,*bk,*bv,*bo; };
struct FfnP  { const float *w1,*b1,*w2,*b2; };
struct EncLayerP { AttnP sa; FfnP ff; const float *ln1_g,*ln1_b,*ln2_g,*ln2_b; };
struct DecLayerP { AttnP sa, ca; FfnP ff;
                   const float *ln1_g,*ln1_b,*ln2_g,*ln2_b,*ln3_g,*ln3_b; };
struct KFParams {
  const float *F0,*H0,*Qm,*z_seq;
  const float *enc_in_w,*enc_in_b,*dec_in_w,*dec_in_b;
  EncLayerP enc[2];
  DecLayerP dec[2];
  const float *enc_norm_g,*enc_norm_b,*dec_norm_g,*dec_norm_b,*gain_w,*gain_b;
  float *out;
};

__device__ __forceinline__ v8f wmma4(v2f a, v2f b, v8f c) {
  // D = A(16x4,f32) * B(4x16,f32) + C(16x16,f32)
  return __builtin_amdgcn_wmma_f32_16x16x4_f32(
      /*neg_a=*/false, a, /*neg_b=*/false, b,
      /*c_mod=*/(short)0, c, /*reuse_a=*/false, /*reuse_b=*/false);
}

// Out(SEQP x 64) = A(SEQP x 64, LDS) @ W(64 x 64, global row-major) + bias
// 8 waves: wave -> (m0 = (wave/4)*16, n0 = (wave%4)*16). Optional ReLU.
__device__ __forceinline__ void gemm_aw(const float* __restrict__ A,
                                        const float* __restrict__ W,
                                        const float* __restrict__ bias,
                                        float* __restrict__ Out,
                                        int wave, int lr, int half, int relu)
{
  const int m0 = (wave >> 2) << 4;
  const int n0 = (wave & 3) << 4;
  v8f c = {0.f,0.f,0.f,0.f,0.f,0.f,0.f,0.f};
  const float* arow = A + (m0 + lr) * Dm + 2 * half;      // A[m][k0+2h .. +1]
  const float* wcol = W + (2 * half) * Dm + n0 + lr;      // W[k0+2h][n]
#pragma unroll
  for (int k0 = 0; k0 < Dm; k0 += 4) {
    v2f a, b;
    a.x = arow[k0];          a.y = arow[k0 + 1];
    b.x = wcol[k0 * Dm];     b.y = wcol[k0 * Dm + Dm];
    c = wmma4(a, b, c);
  }
  const float bb = bias ? bias[n0 + lr] : 0.0f;
  float* orow = Out + (m0 + 8 * half) * Dm + n0 + lr;
#pragma unroll
  for (int i = 0; i < 8; ++i) {
    float v = c[i] + bb;
    if (relu) v = fmaxf(v, 0.0f);
    orow[i * Dm] = v;
  }
}

// S[h](32x32) = Q_h(32x32) @ K_h^T(32x32) * scale
// 8 waves: h = wave/4; tile t = wave%4 -> (m0,n0) in {0,16}x{0,16}
__device__ __forceinline__ void gemm_qkT(const float* __restrict__ Q,
                                         const float* __restrict__ K,
                                         float* __restrict__ S,
                                         int wave, int lr, int half, float scale)
{
  const int h  = wave >> 2;
  const int t  = wave & 3;
  const int m0 = (t >> 1) << 4;
  const int n0 = (t & 1) << 4;
  const int ho = h << 5;
  v8f c = {0.f,0.f,0.f,0.f,0.f,0.f,0.f,0.f};
  const float* qrow = Q + (m0 + lr) * Dm + ho + 2 * half;
  const float* krow = K + (n0 + lr) * Dm + ho + 2 * half; // B[k][n] = K[n][k]
#pragma unroll
  for (int k0 = 0; k0 < 32; k0 += 4) {
    v2f a, b;
    a.x = qrow[k0]; a.y = qrow[k0 + 1];
    b.x = krow[k0]; b.y = krow[k0 + 1];
    c = wmma4(a, b, c);
  }
  float* srow = S + (h << 10) + (m0 + 8 * half) * 32 + n0 + lr;
#pragma unroll
  for (int i = 0; i < 8; ++i) srow[i * 32] = c[i] * scale;
}

// O[:, h*32 + n] = softmax(S_h)(32x32) @ V_h(32x32)
__device__ __forceinline__ void gemm_av(const float* __restrict__ S,
                                        const float* __restrict__ V,
                                        float* __restrict__ O,
                                        int wave, int lr, int half)
{
  const int h  = wave >> 2;
  const int t  = wave & 3;
  const int m0 = (t >> 1) << 4;
  const int n0 = (t & 1) << 4;
  const int ho = h << 5;
  v8f c = {0.f,0.f,0.f,0.f,0.f,0.f,0.f,0.f};
  const float* arow = S + (h << 10) + (m0 + lr) * 32 + 2 * half;
  const float* vcol = V + (2 * half) * Dm + ho + n0 + lr;
#pragma unroll
  for (int k0 = 0; k0 < 32; k0 += 4) {
    v2f a, b;
    a.x = arow[k0];       a.y = arow[k0 + 1];
    b.x = vcol[k0 * Dm];  b.y = vcol[k0 * Dm + Dm];
    c = wmma4(a, b, c);
  }
  float* orow = O + (m0 + 8 * half) * Dm + ho + n0 + lr;
#pragma unroll
  for (int i = 0; i < 8; ++i) orow[i * Dm] = c[i];
}

// masked softmax over keys; valid keys are [lo, WIN). pad keys -> exactly 0.
__device__ __forceinline__ void softmax_rows(float* __restrict__ S, int tid, int lo)
{
  if (tid < 64) {
    float* row = S + ((tid >> 5) << 10) + (tid & 31) * 32;
    float m = -3.0e38f;
    for (int j = 0; j < 32; ++j) {
      float s = (j >= lo && j < WIN) ? row[j] : -1.0e9f;
      row[j] = s;
      m = fmaxf(m, s);
    }
    float sum = 0.f;
    for (int j = 0; j < 32; ++j) { float e = expf(row[j] - m); row[j] = e; sum += e; }
    const float inv = 1.f / sum;
    for (int j = 0; j < 32; ++j) row[j] *= inv;
  }
}

// X = LayerNorm(X + Tm) * g + b, row-wise over 64; one thread per row.
__device__ __forceinline__ void add_ln(float* __restrict__ X, const float* __restrict__ Tm,
                                       const float* __restrict__ g, const float* __restrict__ bt,
                                       int tid)
{
  __syncthreads();
  if (tid < SEQP) {
    float* xr = X + tid * Dm;
    const float* tr = Tm + tid * Dm;
    float mu = 0.f;
    for (int c2 = 0; c2 < Dm; ++c2) mu += xr[c2] + tr[c2];
    mu *= (1.f / Dm);
    float var = 0.f;
    for (int c2 = 0; c2 < Dm; ++c2) { float v = xr[c2] + tr[c2] - mu; var += v * v; }
    var *= (1.f / Dm);
    const float rs = rsqrtf(var + KF_EPS);
    for (int c2 = 0; c2 < Dm; ++c2)
      xr[c2] = (xr[c2] + tr[c2] - mu) * rs * g[c2] + bt[c2];
  }
  __syncthreads();
}

// Multi-head attention; result (after wo projection) left in Kb.
__device__ void mha(const float* Xq, const float* Xkv, const AttnP& p, int lo,
                    float* Qb, float* Kb, float* Vb, float* Sb,
                    int wave, int lr, int half, int tid)
{
  gemm_aw(Xq,  p.wq, p.bq, Qb, wave, lr, half, 0);
  gemm_aw(Xkv, p.wk, p.bk, Kb, wave, lr, half, 0);
  gemm_aw(Xkv, p.wv, p.bv, Vb, wave, lr, half, 0);
  __syncthreads();
  gemm_qkT(Qb, Kb, Sb, wave, lr, half, 0.17677669529663687f); // 1/sqrt(32)
  __syncthreads();
  softmax_rows(Sb, tid, lo);
  __syncthreads();
  gemm_av(Sb, Vb, Qb, wave, lr, half);          // O overwrites Qb
  __syncthreads();
  gemm_aw(Qb, p.wo, p.bo, Kb, wave, lr, half, 0); // result in Kb
}

__global__ __launch_bounds__(256)
void kalmanformer_kernel(KFParams P)
{
  __shared__ float X [SEQP * Dm];   // encoder activations / memory
  __shared__ float Yb[SEQP * Dm];   // decoder activations
  __shared__ float Qb[SEQP * Dm];
  __shared__ float Kb[SEQP * Dm];
  __shared__ float Vb[SEQP * Dm];
  __shared__ float Sb[2 * SEQP * SEQP];
  __shared__ float enc_buf[WIN * 8];
  __shared__ float dec_buf[WIN * 4];
  // st: 0..3 x_est | 4..19 P_est | 20..21 z_prev | 22..25 x_pred |
  //     26..27 z_pred | 28..43 P_pred | 44..51 Kgain | 52..59 feat | 60..61 z_k
  __shared__ float st[64];

  const int tid  = threadIdx.x;
  const int wave = tid >> 5;
  const int lane = tid & 31;
  const int half = lane >> 4;
  const int lr   = lane & 15;
  const int b    = blockIdx.x;

  for (int i = tid; i < WIN * 8; i += 256) enc_buf[i] = 0.f;
  for (int i = tid; i < WIN * 4; i += 256) dec_buf[i] = 0.f;
  if (tid < 64) st[tid] = 0.f;
  __syncthreads();
  if (tid < 16) st[4 + tid] = ((tid & 3) == (tid >> 2)) ? 1.f : 0.f; // P = I
  __syncthreads();

  for (int k = 0; k < Tsz; ++k) {
    const int enc_valid = min(max(k, 1), WIN);
    const int dec_valid = min(k + 1, WIN);
    const int src_lo = WIN - enc_valid;
    const int tgt_lo = WIN - dec_valid;

    // ---- Kalman predict + decoder feature (tiny: one thread) ----
    if (tid == 0) {
      const float zk0 = P.z_seq[(b * Tsz + k) * 2 + 0];
      const float zk1 = P.z_seq[(b * Tsz + k) * 2 + 1];
      float xp[4];
      for (int i = 0; i < 4; ++i) {
        float s = 0.f;
        for (int j = 0; j < 4; ++j) s += P.F0[i * 4 + j] * st[j];
        xp[i] = s; st[22 + i] = s;
      }
      float zp[2];
      for (int m = 0; m < 2; ++m) {
        float s = 0.f;
        for (int j = 0; j < 4; ++j) s += P.H0[m * 4 + j] * xp[j];
        zp[m] = s; st[26 + m] = s;
      }
      float FP[16];
      for (int i = 0; i < 4; ++i)
        for (int j = 0; j < 4; ++j) {
          float s = 0.f;
          for (int l = 0; l < 4; ++l) s += P.F0[i * 4 + l] * st[4 + l * 4 + j];
          FP[i * 4 + j] = s;
        }
      for (int i = 0; i < 4; ++i)
        for (int j = 0; j < 4; ++j) {
          float s = 0.f;
          for (int l = 0; l < 4; ++l) s += FP[i * 4 + l] * P.F0[j * 4 + l];
          st[28 + i * 4 + j] = s + P.Qm[i * 4 + j];
        }
      // dec_feat = LN4([z_k - z_prev, z_k - z_pred])
      float f[4] = { zk0 - st[20], zk1 - st[21], zk0 - zp[0], zk1 - zp[1] };
      float mu = 0.25f * (f[0] + f[1] + f[2] + f[3]);
      float var = 0.f;
      for (int i = 0; i < 4; ++i) { float v = f[i] - mu; var += v * v; }
      var *= 0.25f;
      const float rs = rsqrtf(var + KF_EPS);
      for (int i = 0; i < 4; ++i)
        st[52 + i] = (f[i] - mu) * rs * P.dec_norm_g[i] + P.dec_norm_b[i];
      st[60] = zk0; st[61] = zk1;
    }
    __syncthreads();

    // ---- dec_buf shift + append ----
    {
      float t0 = (tid < (WIN - 1) * 4) ? dec_buf[tid + 4] : 0.f;
      __syncthreads();
      if (tid < (WIN - 1) * 4)      dec_buf[tid] = t0;
      else if (tid < WIN * 4)       dec_buf[tid] = st[52 + tid - (WIN - 1) * 4];
      __syncthreads();
    }

    // ---- input projections (K=8 / K=4: scalar) ----
    for (int idx = tid; idx < SEQP * Dm; idx += 256) {
      const int r = idx >> 6, c2 = idx & 63;
      float se = 0.f, sd = 0.f;
      if (r < WIN) {
        se = P.enc_in_b[c2];
        for (int j = 0; j < 8; ++j) se += enc_buf[r * 8 + j] * P.enc_in_w[j * Dm + c2];
        sd = P.dec_in_b[c2];
        for (int j = 0; j < 4; ++j) sd += dec_buf[r * 4 + j] * P.dec_in_w[j * Dm + c2];
      }
      X[idx] = se; Yb[idx] = sd;
    }
    __syncthreads();

    // ---- encoder ----
    for (int l = 0; l < 2; ++l) {
      const EncLayerP& e = P.enc[l];
      mha(X, X, e.sa, src_lo, Qb, Kb, Vb, Sb, wave, lr, half, tid);
      add_ln(X, Kb, e.ln1_g, e.ln1_b, tid);
      gemm_aw(X,  e.ff.w1, e.ff.b1, Qb, wave, lr, half, 1);
      __syncthreads();
      gemm_aw(Qb, e.ff.w2, e.ff.b2, Kb, wave, lr, half, 0);
      add_ln(X, Kb, e.ln2_g, e.ln2_b, tid);
    }

    // ---- decoder (X holds encoder memory, preserved) ----
    for (int l = 0; l < 2; ++l) {
      const DecLayerP& d = P.dec[l];
      mha(Yb, Yb, d.sa, tgt_lo, Qb, Kb, Vb, Sb, wave, lr, half, tid);
      add_ln(Yb, Kb, d.ln1_g, d.ln1_b, tid);
      mha(Yb, X, d.ca, src_lo, Qb, Kb, Vb, Sb, wave, lr, half, tid);
      add_ln(Yb, Kb, d.ln2_g, d.ln2_b, tid);
      gemm_aw(Yb, d.ff.w1, d.ff.b1, Qb, wave, lr, half, 1);
      __syncthreads();
      gemm_aw(Qb, d.ff.w2, d.ff.b2, Kb, wave, lr, half, 0);
      add_ln(Yb, Kb, d.ln3_g, d.ln3_b, tid);
    }

    // ---- gain head: K = last @ gain_w + gain_b ----
    if (tid < 8) {
      float s = P.gain_b[tid];
      const float* last = Yb + (WIN - 1) * Dm;
      for (int c2 = 0; c2 < Dm; ++c2) s += last[c2] * P.gain_w[c2 * 8 + tid];
      st[44 + tid] = s;   // K[i][m] = st[44 + i*2 + m]
    }
    __syncthreads();

    // ---- Kalman update + enc feature + output ----
    if (tid == 0) {
      const float zk0 = st[60], zk1 = st[61];
      const float y0 = zk0 - st[26], y1 = zk1 - st[27];
      float xn[4];
      for (int i = 0; i < 4; ++i)
        xn[i] = st[22 + i] + st[44 + i * 2] * y0 + st[44 + i * 2 + 1] * y1;
      float A[16];
      for (int i = 0; i < 4; ++i)
        for (int j = 0; j < 4; ++j)
          A[i * 4 + j] = ((i == j) ? 1.f : 0.f)
                         - (st[44 + i * 2] * P.H0[j] + st[44 + i * 2 + 1] * P.H0[4 + j]);
      float Pn[16];
      for (int i = 0; i < 4; ++i)
        for (int j = 0; j < 4; ++j) {
          float s = 0.f;
          for (int l = 0; l < 4; ++l) s += A[i * 4 + l] * st[28 + l * 4 + j];
          Pn[i * 4 + j] = s;
        }
      // enc_feat = LN8([x_new - x_prev, x_new - x_pred])  (x_prev == old x_est)
      float f[8];
      for (int i = 0; i < 4; ++i) { f[i] = xn[i] - st[i]; f[4 + i] = xn[i] - st[22 + i]; }
      float mu = 0.f;
      for (int i = 0; i < 8; ++i) mu += f[i];
      mu *= 0.125f;
      float var = 0.f;
      for (int i = 0; i < 8; ++i) { float v = f[i] - mu; var += v * v; }
      var *= 0.125f;
      const float rs = rsqrtf(var + KF_EPS);
      for (int i = 0; i < 8; ++i)
        st[52 + i] = (f[i] - mu) * rs * P.enc_norm_g[i] + P.enc_norm_b[i];
      // commit state + output
      for (int i = 0; i < 4; ++i) {
        st[i] = xn[i];
        P.out[(b * Tsz + k) * 4 + i] = xn[i];
      }
      for (int i = 0; i < 16; ++i) st[4 + i] = Pn[i];
      st[20] = zk0; st[21] = zk1;
    }
    __syncthreads();

    // ---- enc_buf shift + append ----
    {
      float t0 = (tid < (WIN - 1) * 8) ? enc_buf[tid + 8] : 0.f;
      __syncthreads();
      if (tid < (WIN - 1) * 8)      enc_buf[tid] = t0;
      else if (tid < WIN * 8)       enc_buf[tid] = st[52 + tid - (WIN - 1) * 8];
      __syncthreads();
    }
  }
}

// ---------------------------------------------------------------------------
// Host launch. Input order assumes pytree (sorted-key) flattening of
// setup_inputs():
//  0:F0 1:H0 2:Q 3:R
//  4:dec_norm_b 5:dec_norm_g 6:enc_norm_b 7:enc_norm_g 8:gain_b 9:gain_w
//  10:tf.dec_in_b 11:tf.dec_in_w
//  12..37 dec_layer0: ca{bk,bo,bq,bv,wk,wo,wq,wv} ff{b1,b2,w1,w2}
//                     ln1_b ln1_g ln2_b ln2_g ln3_b ln3_g
//                     sa{bk,bo,bq,bv,wk,wo,wq,wv}
//  38..63 dec_layer1 (same layout)
//  64:tf.enc_in_b 65:tf.enc_in_w
//  66..81 enc_layer0: ff{b1,b2,w1,w2} ln1_b ln1_g ln2_b ln2_g
//                     sa{bk,bo,bq,bv,wk,wo,wq,wv}
//  82..97 enc_layer1
//  98:z_seq
// ---------------------------------------------------------------------------
extern "C" void kernel_launch(void* const* d_in, const int* in_sizes, int n_in,
                              void* d_out, int out_size, void* d_ws, size_t ws_size,
                              hipStream_t stream) {
  (void)in_sizes; (void)n_in; (void)out_size; (void)d_ws; (void)ws_size;
  auto f = [&](int i) { return (const float*)d_in[i]; };

  KFParams P;
  P.F0 = f(0); P.H0 = f(1); P.Qm = f(2);
  P.dec_norm_b = f(4); P.dec_norm_g = f(5);
  P.enc_norm_b = f(6); P.enc_norm_g = f(7);
  P.gain_b = f(8); P.gain_w = f(9);
  P.dec_in_b = f(10); P.dec_in_w = f(11);
  int idx = 12;
  for (int l = 0; l < 2; ++l) {
    DecLayerP& d = P.dec[l];
    d.ca.bk = f(idx + 0);  d.ca.bo = f(idx + 1);  d.ca.bq = f(idx + 2);  d.ca.bv = f(idx + 3);
    d.ca.wk = f(idx + 4);  d.ca.wo = f(idx + 5);  d.ca.wq = f(idx + 6);  d.ca.wv = f(idx + 7);
    d.ff.b1 = f(idx + 8);  d.ff.b2 = f(idx + 9);  d.ff.w1 = f(idx + 10); d.ff.w2 = f(idx + 11);
    d.ln1_b = f(idx + 12); d.ln1_g = f(idx + 13);
    d.ln2_b = f(idx + 14); d.ln2_g = f(idx + 15);
    d.ln3_b = f(idx + 16); d.ln3_g = f(idx + 17);
    d.sa.bk = f(idx + 18); d.sa.bo = f(idx + 19); d.sa.bq = f(idx + 20); d.sa.bv = f(idx + 21);
    d.sa.wk = f(idx + 22); d.sa.wo = f(idx + 23); d.sa.wq = f(idx + 24); d.sa.wv = f(idx + 25);
    idx += 26;
  }
  P.enc_in_b = f(64); P.enc_in_w = f(65);
  idx = 66;
  for (int l = 0; l < 2; ++l) {
    EncLayerP& e = P.enc[l];
    e.ff.b1 = f(idx + 0);  e.ff.b2 = f(idx + 1);  e.ff.w1 = f(idx + 2);  e.ff.w2 = f(idx + 3);
    e.ln1_b = f(idx + 4);  e.ln1_g = f(idx + 5);
    e.ln2_b = f(idx + 6);  e.ln2_g = f(idx + 7);
    e.sa.bk = f(idx + 8);  e.sa.bo = f(idx + 9);  e.sa.bq = f(idx + 10); e.sa.bv = f(idx + 11);
    e.sa.wk = f(idx + 12); e.sa.wo = f(idx + 13); e.sa.wq = f(idx + 14); e.sa.wv = f(idx + 15);
    idx += 16;
  }
  P.z_seq = f(98);
  P.out = (float*)d_out;

  kalmanformer_kernel<<<Bsz, 256, 0, stream>>>(P);
}